// CompressedInteractionNetwork_9105330667837
// MI455X (gfx1250) — compile-verified
//
#include <hip/hip_runtime.h>

typedef __attribute__((ext_vector_type(2))) float v2f;
typedef __attribute__((ext_vector_type(8))) float v8f;

#define BATCH 2048
#define FIELD 32
#define EMB   64
#define O_DIM 64

// ---------------------------------------------------------------------------
// Pack W[o, h, m] (m fastest) into per-WMMA-fragment order so the layer
// kernel's A-operand is one coalesced global_load_b64 per v_wmma.
// Fragment (ot, ks) covers o in [16*ot, 16*ot+16), k = 4*ks .. 4*ks+3 with
// k = h*32 + m.  Lane L holds A[L%16, 2*(L/16)+j] for j=0,1 (ISA 16x4 layout).
// ---------------------------------------------------------------------------
__global__ void pack_w_kernel(const float* __restrict__ W, float* __restrict__ Wp,
                              int H, int total) {
  int p = blockIdx.x * blockDim.x + threadIdx.x;
  if (p >= total) return;
  int j    = p & 1;
  int L    = (p >> 1) & 31;
  int frag = p >> 6;
  int ks   = frag % (H * 8);
  int ot   = frag / (H * 8);
  int h    = ks >> 3;
  int m    = ((ks & 7) << 2) + ((L >> 4) << 1) + j;
  int o    = (ot << 4) + (L & 15);
  Wp[p] = W[(o * H + h) * FIELD + m];
}

// ---------------------------------------------------------------------------
// One CIN layer: h[b,o,d] = sum_{h,m} W[o,h,m] * xk[b,h,d] * x0[b,m,d] + bias[o]
// Block = one batch row b. 16 waves: wave w -> (o-tile = w&3, d-tile = w>>2).
// Each wave runs a K-loop of H*8 v_wmma_f32_16x16x4_f32.
//   A (16 o  x 4 k): packed weights, coalesced b64 load, L2-resident.
//   B (4 k  x 16 d): built in-registers: xv = xk[h,d] (broadcast) times two
//                    x0[m,d] LDS reads (stride-80 rows -> conflict-free).
// POOL=true fuses the sum over d (layer 3) so h3 is never materialized.
// ---------------------------------------------------------------------------
template <int H, bool SAME_X, bool POOL>
__global__ __launch_bounds__(512)
void cin_layer_kernel(const float* __restrict__ x0g,
                      const float* __restrict__ xkg,
                      const float* __restrict__ Wp,
                      const float* __restrict__ bias,
                      float* __restrict__ outg) {
  __shared__ float x0s[FIELD][80];
  __shared__ float xks[SAME_X ? 1 : H][80];
  __shared__ float pool[POOL ? O_DIM : 1][4];

  const int b    = blockIdx.x;
  const int tid  = threadIdx.x;
  const int w    = tid >> 5;
  const int L    = tid & 31;
  const int n    = L & 15;       // d within tile / column of C
  const int half = L >> 4;       // 0: K rows {0,1}, 1: K rows {2,3}
  const int ot   = w & 3;
  const int dt   = w >> 2;
  const int d    = (dt << 4) + n;
  const int o0   = ot << 4;

  for (int idx = tid; idx < FIELD * EMB; idx += 512)
    x0s[idx >> 6][idx & 63] = x0g[(size_t)b * FIELD * EMB + idx];
  if constexpr (!SAME_X) {
    for (int idx = tid; idx < H * EMB; idx += 512)
      xks[idx >> 6][idx & 63] = xkg[(size_t)b * H * EMB + idx];
  }
  __syncthreads();

  v8f acc = {};
  const float* wbase = Wp + (size_t)(ot * (H * 8)) * 64 + (L << 1);
  for (int h = 0; h < H; ++h) {
    const float xv = SAME_X ? x0s[h][d] : xks[h][d];
    const float* wph = wbase + (size_t)h * 8 * 64;
#pragma unroll
    for (int mq = 0; mq < 8; ++mq) {
      const v2f a = *(const v2f*)(wph + mq * 64);      // coalesced, L2-hot
      const int m = (mq << 2) + (half << 1);
      v2f bb;
      bb.x = xv * x0s[m][d];
      bb.y = xv * x0s[m + 1][d];
      acc = __builtin_amdgcn_wmma_f32_16x16x4_f32(
          false, a, false, bb, (short)0, acc, false, false);
    }
  }

  if constexpr (!POOL) {
#pragma unroll
    for (int i = 0; i < 8; ++i) {
      const int o = o0 + i + (half << 3);
      outg[((size_t)b * O_DIM + o) * EMB + d] = acc[i] + bias[o];
    }
  } else {
    // Reduce each C row over its 16 d-columns (within each 16-lane group).
#pragma unroll
    for (int i = 0; i < 8; ++i) {
      float v = acc[i];
      v += __shfl_xor(v, 1, 16);
      v += __shfl_xor(v, 2, 16);
      v += __shfl_xor(v, 4, 16);
      v += __shfl_xor(v, 8, 16);
      if (n == 0) pool[o0 + i + (half << 3)][dt] = v;
    }
    __syncthreads();
    if (tid < O_DIM) {
      float s = pool[tid][0] + pool[tid][1] + pool[tid][2] + pool[tid][3];
      // sum_d (tile + bias) = tile_sum + EMB * bias
      outg[(size_t)b * O_DIM + tid] = s + 64.0f * bias[tid];
    }
  }
}

// ---------------------------------------------------------------------------
// out[b] = b_out + sum_o Wout[o]*sum_d h1 + Wout[64+o]*sum_d h2
//                 + Wout[128+o]*pooled3[b,o]
// ---------------------------------------------------------------------------
__global__ __launch_bounds__(256)
void cin_final_kernel(const float* __restrict__ h1, const float* __restrict__ h2,
                      const float* __restrict__ pooled3,
                      const float* __restrict__ Wout, const float* __restrict__ bout,
                      float* __restrict__ out) {
  __shared__ float red[256];
  const int b = blockIdx.x, t = threadIdx.x;
  const int o = t >> 2, dq = t & 3;
  const float* p1 = h1 + ((size_t)b * O_DIM + o) * EMB + dq * 16;
  const float* p2 = h2 + ((size_t)b * O_DIM + o) * EMB + dq * 16;
  float s1 = 0.f, s2 = 0.f;
#pragma unroll
  for (int i = 0; i < 16; ++i) { s1 += p1[i]; s2 += p2[i]; }
  float partial = Wout[o] * s1 + Wout[64 + o] * s2;
  if (dq == 0) partial += Wout[128 + o] * pooled3[(size_t)b * O_DIM + o];
  red[t] = partial;
  __syncthreads();
  for (int s = 128; s > 0; s >>= 1) {
    if (t < s) red[t] += red[t + s];
    __syncthreads();
  }
  if (t == 0) out[b] = red[0] + bout[0];
}

extern "C" void kernel_launch(void* const* d_in, const int* in_sizes, int n_in,
                              void* d_out, int out_size, void* d_ws, size_t ws_size,
                              hipStream_t stream) {
  (void)in_sizes; (void)n_in; (void)out_size; (void)ws_size;
  const float* x0   = (const float*)d_in[0];
  const float* W1   = (const float*)d_in[1];
  const float* b1   = (const float*)d_in[2];
  const float* W2   = (const float*)d_in[3];
  const float* b2   = (const float*)d_in[4];
  const float* W3   = (const float*)d_in[5];
  const float* b3   = (const float*)d_in[6];
  const float* Wout = (const float*)d_in[7];
  const float* bout = (const float*)d_in[8];

  float* ws    = (float*)d_ws;
  float* h1    = ws;                 // 2048*64*64           = 8,388,608 f
  float* h2    = h1 + 8388608;       //                        8,388,608 f
  float* wp1   = h2 + 8388608;       // 64*32*32             =    65,536 f
  float* wp2   = wp1 + 65536;        // 64*64*32             =   131,072 f
  float* wp3   = wp2 + 131072;       //                         131,072 f
  float* pool3 = wp3 + 131072;       // 2048*64              =   131,072 f
  // total ~68.9 MB of workspace

  pack_w_kernel<<<65536 / 256, 256, 0, stream>>>(W1, wp1, 32, 65536);
  pack_w_kernel<<<131072 / 256, 256, 0, stream>>>(W2, wp2, 64, 131072);
  pack_w_kernel<<<131072 / 256, 256, 0, stream>>>(W3, wp3, 64, 131072);

  cin_layer_kernel<32, true,  false><<<BATCH, 512, 0, stream>>>(x0, nullptr, wp1, b1, h1);
  cin_layer_kernel<64, false, false><<<BATCH, 512, 0, stream>>>(x0, h1,      wp2, b2, h2);
  cin_layer_kernel<64, false, true ><<<BATCH, 512, 0, stream>>>(x0, h2,      wp3, b3, pool3);

  cin_final_kernel<<<BATCH, 256, 0, stream>>>(h1, h2, pool3, Wout, bout, (float*)d_out);
}